// CrackBinaryFilter_90821378441815
// MI455X (gfx1250) — compile-verified
//
#include <hip/hip_runtime.h>
#include <math.h>
#include <stdint.h>

// ---------------- constants ----------------
static constexpr int IMG_H = 4096;
static constexpr int IMG_W = 4096;
static constexpr int NPIX  = IMG_H * IMG_W;
static constexpr int NBINS = 65536;
static constexpr unsigned GMAX = 1u << 20;   // gather capacity (floats)

// workspace byte offsets
static constexpr size_t OFF_BLUR   = 0;                          // float[NPIX]      64 MB
static constexpr size_t OFF_HIST   = 67108864;                   // uint[NBINS]     256 KB
static constexpr size_t OFF_CTRL   = OFF_HIST + 262144;          // uint[64]        256 B
static constexpr size_t OFF_GATHER = OFF_CTRL + 256;             // float[GMAX]       4 MB
static constexpr size_t OFF_ERODED = OFF_GATHER + 4194304;       // u8[NPIX]         16 MB

typedef __attribute__((ext_vector_type(2))) float v2f;
typedef __attribute__((ext_vector_type(8))) float v8f;

#if __has_builtin(__builtin_amdgcn_wmma_f32_16x16x4_f32)
#define HAS_WMMA_F32 1
#endif
#if __has_builtin(__builtin_amdgcn_global_load_async_to_lds_b32)
#define HAS_ASYNC_LDS 1
#endif

__device__ __forceinline__ float bandw(int d, float w0, float w1) {
  // 3-tap band: d==0 or 2 -> w0 ; d==1 -> w1 ; else 0
  return (d == 1) ? w1 : ((d == 0 || d == 2) ? w0 : 0.0f);
}

__device__ __forceinline__ int binOf(float v) {
  int b = (int)(v * 65536.0f);
  b = b < 0 ? 0 : b;
  return b > (NBINS - 1) ? (NBINS - 1) : b;
}

// ---------------- zero scratch (hist + ctrl) ----------------
__global__ void k_zero(unsigned* __restrict__ p, int n) {
  int i = blockIdx.x * blockDim.x + threadIdx.x;
  if (i < n) p[i] = 0u;
}

// ---------------- K1: grayscale + separable gaussian via WMMA + histogram ----
// Block: 128 threads (4 waves), 32x32 output tile. Each wave owns a 16x16
// sub-tile and runs the blur as two banded 16x16x20 f32 matmuls on the matrix
// pipe (K padded to 20 with zero weights; LDS addresses clamped so padding
// reads finite data -> weight 0 kills it).
__global__ __launch_bounds__(128) void k_gray_blur_hist(
    const float* __restrict__ img, float* __restrict__ blur,
    unsigned* __restrict__ hist, float w0, float w1) {
  __shared__ float Gs[34][36];   // gray tile with 1-px halo (region rows -1..32)
  const int tid = threadIdx.x;
  const int bx = blockIdx.x, by = blockIdx.y;
  const float* Rp = img;
  const float* Gp = img + NPIX;
  const float* Bp = img + 2 * NPIX;

  for (int i = tid; i < 34 * 34; i += 128) {
    int r = i / 34, c = i % 34;
    int gy = by * 32 + r - 1;
    int gx = bx * 32 + c - 1;
    gy = gy < 0 ? -gy : (gy > IMG_H - 1 ? 2 * (IMG_H - 1) - gy : gy);  // reflect
    gx = gx < 0 ? -gx : (gx > IMG_W - 1 ? 2 * (IMG_W - 1) - gx : gx);
    long idx = (long)gy * IMG_W + gx;
    Gs[r][c] = fmaf(0.2989f, Rp[idx], fmaf(0.587f, Gp[idx], 0.114f * Bp[idx]));
  }
  __syncthreads();

  const int wave = tid >> 5;
  const int lane = tid & 31;
  const int n    = lane & 15;    // A-row (M) / B-col (N) index for this lane
  const int hi   = lane >> 4;    // 0: K pair {0,1}; 1: K pair {2,3}
  const int khalf = hi * 2;
  const int wrow = (wave & 2) * 8;    // 0 or 16
  const int wcol = (wave & 1) * 16;   // 0 or 16

#if defined(HAS_WMMA_F32)
  __shared__ float HV[4][16][20];  // per-wave vertical-pass result, cols j=0..17

  // ---- vertical pass: Hv(16x18) = Wv(16x20-band) x Gs(20x{16,16}) ----
  v8f accA = {0, 0, 0, 0, 0, 0, 0, 0};  // cols j = 0..15
  v8f accB = {0, 0, 0, 0, 0, 0, 0, 0};  // cols j = 2..17 (only 16,17 kept)
#pragma unroll
  for (int c4 = 0; c4 < 5; ++c4) {
    int k0 = 4 * c4 + khalf;
    int k1 = k0 + 1;
    v2f a;
    a.x = bandw(k0 - n, w0, w1);   // Wv[m=n][k]
    a.y = bandw(k1 - n, w0, w1);
    int r0 = wrow + k0; r0 = r0 > 33 ? 33 : r0;   // clamp into region (weight 0 beyond)
    int r1 = wrow + k1; r1 = r1 > 33 ? 33 : r1;
    v2f bA, bB;
    bA.x = Gs[r0][wcol + n];
    bA.y = Gs[r1][wcol + n];
    bB.x = Gs[r0][wcol + n + 2];
    bB.y = Gs[r1][wcol + n + 2];
    accA = __builtin_amdgcn_wmma_f32_16x16x4_f32(false, a, false, bA, (short)0, accA, false, false);
    accB = __builtin_amdgcn_wmma_f32_16x16x4_f32(false, a, false, bB, (short)0, accB, false, false);
  }
#pragma unroll
  for (int r = 0; r < 8; ++r) {
    int row = r + hi * 8;                      // D layout: VGPR r -> M=r (lanes 0-15) / r+8
    HV[wave][row][n] = accA[r];                // j = n
    if (n >= 14) HV[wave][row][n + 2] = accB[r];  // j = 16,17
  }

  // ---- horizontal pass: out(16x16) = Hv(16x20) x Wh(20x16-band) ----
  v8f acc = {0, 0, 0, 0, 0, 0, 0, 0};
#pragma unroll
  for (int c4 = 0; c4 < 5; ++c4) {
    int k0 = 4 * c4 + khalf;
    int k1 = k0 + 1;
    int kc0 = k0 > 17 ? 17 : k0;
    int kc1 = k1 > 17 ? 17 : k1;
    v2f a, b;
    a.x = HV[wave][n][kc0];
    a.y = HV[wave][n][kc1];
    b.x = bandw(k0 - n, w0, w1);   // Wh[k][n]
    b.y = bandw(k1 - n, w0, w1);
    acc = __builtin_amdgcn_wmma_f32_16x16x4_f32(false, a, false, b, (short)0, acc, false, false);
  }
#pragma unroll
  for (int r = 0; r < 8; ++r) {
    int row = wrow + r + hi * 8;
    int col = wcol + n;
    float v = acc[r];
    int gy = by * 32 + row, gx = bx * 32 + col;
    blur[(long)gy * IMG_W + gx] = v;
    atomicAdd(&hist[binOf(v)], 1u);
  }
#else
  // VALU fallback (also what the host-side parse sees)
  for (int i = tid; i < 1024; i += 128) {
    int y = i >> 5, x = i & 31;
    int r = y + 1, c = x + 1;
    float a0 = fmaf(w0, Gs[r - 1][c - 1], fmaf(w1, Gs[r - 1][c], w0 * Gs[r - 1][c + 1]));
    float a1 = fmaf(w0, Gs[r][c - 1],     fmaf(w1, Gs[r][c],     w0 * Gs[r][c + 1]));
    float a2 = fmaf(w0, Gs[r + 1][c - 1], fmaf(w1, Gs[r + 1][c], w0 * Gs[r + 1][c + 1]));
    float v = fmaf(w0, a0, fmaf(w1, a1, w0 * a2));
    int gy = by * 32 + y, gx = bx * 32 + x;
    blur[(long)gy * IMG_W + gx] = v;
    atomicAdd(&hist[binOf(v)], 1u);
  }
  (void)wave; (void)khalf; (void)wrow; (void)wcol; (void)hi; (void)n;
#endif
}

// ---------------- K2: locate quantile bins -------------------------------
// ctrl[0]=gatherCount ctrl[1]=binLo ctrl[2]=binHi ctrl[3]=countBelow ctrl[4]=threshold bits
__global__ __launch_bounds__(256) void k_scan(const unsigned* __restrict__ hist,
                                              unsigned* __restrict__ ctrl, unsigned i0) {
  __shared__ unsigned partial[256];
  __shared__ unsigned pre[256];
  const int t = threadIdx.x;
  unsigned s = 0;
  for (int b = 0; b < 256; ++b) s += hist[t * 256 + b];
  partial[t] = s;
  __syncthreads();
  if (t == 0) {
    unsigned run = 0;
    for (int i = 0; i < 256; ++i) { pre[i] = run; run += partial[i]; }
  }
  __syncthreads();
  unsigned cum = pre[t];
  const unsigned i1 = i0 + 1;
  for (int b = 0; b < 256; ++b) {
    unsigned c = hist[t * 256 + b];
    unsigned nxt = cum + c;
    if (cum <= i0 && i0 < nxt) { ctrl[1] = (unsigned)(t * 256 + b); ctrl[3] = cum; }
    if (cum <= i1 && i1 < nxt) { ctrl[2] = (unsigned)(t * 256 + b); }
    cum = nxt;
  }
}

// ---------------- K3: gather candidate-bin values ------------------------
__global__ __launch_bounds__(256) void k_gather(const float* __restrict__ blur,
                                                unsigned* __restrict__ ctrl,
                                                float* __restrict__ gat, int n) {
  const unsigned lo = ctrl[1], hiB = ctrl[2];
  const int stride = gridDim.x * blockDim.x;
  for (int i = blockIdx.x * blockDim.x + threadIdx.x; i < n; i += stride) {
    float v = blur[i];
    unsigned b = (unsigned)binOf(v);
    if (b >= lo && b <= hiB) {
      unsigned k = atomicAdd(&ctrl[0], 1u);
      if (k < GMAX) gat[k] = v;
    }
  }
}

// ---------------- K4: exact rank selection -> threshold ------------------
__global__ __launch_bounds__(256) void k_select(const float* __restrict__ gat,
                                                unsigned* __restrict__ ctrl,
                                                unsigned i0, float frac) {
  __shared__ float s0, s1;
  unsigned m = ctrl[0];
  if (m > GMAX) m = GMAX;
  const unsigned below = ctrl[3];
  const unsigned k0 = i0 - below, k1 = k0 + 1;
  if (m == 0 || m > 100000u) {  // degenerate/overflow safety: approx by bin edge
    if (threadIdx.x == 0) ((float*)ctrl)[4] = (float)ctrl[1] / 65536.0f;
    return;
  }
  if (threadIdx.x == 0) { s0 = 0.0f; s1 = 0.0f; }
  __syncthreads();
  for (unsigned i = threadIdx.x; i < m; i += blockDim.x) {
    float x = gat[i];
    unsigned less = 0, eq = 0;
    for (unsigned j = 0; j < m; ++j) {
      float y = gat[j];
      less += (y < x);
      eq += (y == x);
    }
    if (less <= k0 && k0 < less + eq) s0 = x;  // duplicates write same value
    if (less <= k1 && k1 < less + eq) s1 = x;
  }
  __syncthreads();
  if (threadIdx.x == 0) ((float*)ctrl)[4] = s0 + frac * (s1 - s0);
}

// ---------------- K5: mask + 5x5 erosion (zero border) -------------------
__global__ __launch_bounds__(256) void k_erode(const float* __restrict__ blur,
                                               const unsigned* __restrict__ ctrl,
                                               unsigned char* __restrict__ eroded) {
  __shared__ float BT[36 * 37];        // blur tile with 2-px halo
  __shared__ unsigned char RA[36 * 32];
  const int tid = threadIdx.x;
  const int bx = blockIdx.x, by = blockIdx.y;
  const float thr = ((const float*)ctrl)[4];

  for (int i = tid; i < 36 * 36; i += 256) {
    int r = i / 36, c = i % 36;
    int gy = by * 32 + r - 2, gx = bx * 32 + c - 2;
    bool ok = (unsigned)gy < (unsigned)IMG_H && (unsigned)gx < (unsigned)IMG_W;
#if defined(HAS_ASYNC_LDS)
    if (ok) {
      __builtin_amdgcn_global_load_async_to_lds_b32(
          (__attribute__((address_space(1))) int*)(blur + (long)gy * IMG_W + gx),
          (__attribute__((address_space(3))) int*)&BT[r * 37 + c], 0, 0);
    } else {
      BT[r * 37 + c] = -1e30f;  // outside image -> mask 0
    }
#else
    BT[r * 37 + c] = ok ? blur[(long)gy * IMG_W + gx] : -1e30f;
#endif
  }
#if defined(HAS_ASYNC_LDS)
#if __has_builtin(__builtin_amdgcn_s_wait_asynccnt)
  __builtin_amdgcn_s_wait_asynccnt(0);
#else
  asm volatile("s_wait_asynccnt 0" ::: "memory");
#endif
#endif
  __syncthreads();

  for (int i = tid; i < 36 * 32; i += 256) {  // horizontal 5-wide AND
    int r = i / 32, x = i % 32;
    const float* p = &BT[r * 37 + x];
    RA[i] = (unsigned char)((p[0] >= thr) & (p[1] >= thr) & (p[2] >= thr) &
                            (p[3] >= thr) & (p[4] >= thr));
  }
  __syncthreads();
  for (int i = tid; i < 1024; i += 256) {     // vertical 5-tall AND
    int y = i >> 5, x = i & 31;
    unsigned char e = RA[y * 32 + x] & RA[(y + 1) * 32 + x] & RA[(y + 2) * 32 + x] &
                      RA[(y + 3) * 32 + x] & RA[(y + 4) * 32 + x];
    eroded[(long)(by * 32 + y) * IMG_W + (bx * 32 + x)] = e;
  }
}

// ---------------- K6: 5x5 dilation (SAME) -> int32 out -------------------
__global__ __launch_bounds__(256) void k_dilate(const unsigned char* __restrict__ eroded,
                                                int* __restrict__ out) {
  __shared__ unsigned char E[36 * 40];
  __shared__ unsigned char RO[36 * 32];
  const int tid = threadIdx.x;
  const int bx = blockIdx.x, by = blockIdx.y;

  for (int i = tid; i < 36 * 36; i += 256) {
    int r = i / 36, c = i % 36;
    int gy = by * 32 + r - 2, gx = bx * 32 + c - 2;
    bool ok = (unsigned)gy < (unsigned)IMG_H && (unsigned)gx < (unsigned)IMG_W;
    E[r * 40 + c] = ok ? eroded[(long)gy * IMG_W + gx] : (unsigned char)0;
  }
  __syncthreads();
  for (int i = tid; i < 36 * 32; i += 256) {
    int r = i / 32, x = i % 32;
    const unsigned char* p = &E[r * 40 + x];
    RO[i] = (unsigned char)(p[0] | p[1] | p[2] | p[3] | p[4]);
  }
  __syncthreads();
  for (int i = tid; i < 1024; i += 256) {
    int y = i >> 5, x = i & 31;
    unsigned char v = RO[y * 32 + x] | RO[(y + 1) * 32 + x] | RO[(y + 2) * 32 + x] |
                      RO[(y + 3) * 32 + x] | RO[(y + 4) * 32 + x];
    out[(long)(by * 32 + y) * IMG_W + (bx * 32 + x)] = (int)v;
  }
}

// ---------------- launch ----------------
extern "C" void kernel_launch(void* const* d_in, const int* in_sizes, int n_in,
                              void* d_out, int out_size, void* d_ws, size_t ws_size,
                              hipStream_t stream) {
  (void)in_sizes; (void)n_in; (void)out_size; (void)ws_size;
  const float* img = (const float*)d_in[0];
  int* out = (int*)d_out;
  char* ws = (char*)d_ws;
  float*         blur   = (float*)(ws + OFF_BLUR);
  unsigned*      hist   = (unsigned*)(ws + OFF_HIST);
  unsigned*      ctrl   = (unsigned*)(ws + OFF_CTRL);
  float*         gat    = (float*)(ws + OFF_GATHER);
  unsigned char* eroded = (unsigned char*)(ws + OFF_ERODED);

  // gaussian weights for ksize=3, sigma=0.8 (torchvision default)
  double kk = exp(-0.5 * (1.0 / 0.8) * (1.0 / 0.8));
  float w1 = (float)(1.0 / (2.0 * kk + 1.0));
  float w0 = (float)(kk / (2.0 * kk + 1.0));

  // quantile position (linear interpolation, matches jnp.quantile)
  double q = (100.0 - 1.3) / 100.0;
  double pos = q * (double)(NPIX - 1);
  unsigned i0 = (unsigned)pos;
  float frac = (float)(pos - (double)i0);

  const int zn = (262144 + 256) / 4;  // hist + ctrl words (contiguous)
  k_zero<<<(zn + 255) / 256, 256, 0, stream>>>(hist, zn);

  dim3 grid(IMG_W / 32, IMG_H / 32);
  k_gray_blur_hist<<<grid, 128, 0, stream>>>(img, blur, hist, w0, w1);
  k_scan<<<1, 256, 0, stream>>>(hist, ctrl, i0);
  k_gather<<<4096, 256, 0, stream>>>(blur, ctrl, gat, NPIX);
  k_select<<<1, 256, 0, stream>>>(gat, ctrl, i0, frac);
  k_erode<<<grid, 256, 0, stream>>>(blur, ctrl, eroded);
  k_dilate<<<grid, 256, 0, stream>>>(eroded, out);
}